// BNN_SVGD_85985245266033
// MI455X (gfx1250) — compile-verified
//
#include <hip/hip_runtime.h>
#include <hip/hip_bf16.h>

#define N_PART 64
#define HID 1024
#define DIN 16
#define BATCH 2048
#define D_FLAT 18433
#define D_PAD  18448           // next multiple of 16
#define OFF_B1 16384
#define OFF_W2 17408
#define OFF_B2 18432
#define NTILES (D_PAD / 16)    // 1153
#define STEP_OVER_N (0.1f / 64.0f)
#define HT_STRIDE 272          // 16 rows * 17 (padded) per 16x16 h tile

typedef __attribute__((ext_vector_type(2))) float v2f;
typedef __attribute__((ext_vector_type(8))) float v8f;

__device__ __forceinline__ v8f wmma4(v2f a, v2f b, v8f c) {
  // D = A(16x4 f32) * B(4x16 f32) + C(16x16 f32)
  return __builtin_amdgcn_wmma_f32_16x16x4_f32(false, a, false, b, (short)0, c,
                                               false, false);
}

__device__ __forceinline__ v8f vzero8() {
  v8f z;
  for (int i = 0; i < 8; ++i) z[i] = 0.0f;
  return z;
}

// ---------------------------------------------------------------------------
// K0: pack theta (zero-padded to D_PAD) from the four parameter tensors.
// ---------------------------------------------------------------------------
__global__ void pack_theta(const float* __restrict__ W1, const float* __restrict__ b1,
                           const float* __restrict__ W2, const float* __restrict__ b2,
                           float* __restrict__ theta) {
  int idx = blockIdx.x * blockDim.x + threadIdx.x;
  const int total = N_PART * D_PAD;
  for (; idx < total; idx += gridDim.x * blockDim.x) {
    int p = idx / D_PAD, c = idx - p * D_PAD;
    float v;
    if (c < OFF_B1)       v = W1[(size_t)p * (HID * DIN) + c];
    else if (c < OFF_W2)  v = b1[(size_t)p * HID + (c - OFF_B1)];
    else if (c < OFF_B2)  v = W2[(size_t)p * HID + (c - OFF_W2)];
    else if (c == OFF_B2) v = b2[p];
    else                  v = 0.0f;
    theta[idx] = v;
  }
}

// ---------------------------------------------------------------------------
// K1: per-particle forward + backward -> dlog (padded). 1 block = 1 particle.
// 8 waves; wave w owns HID columns [w*128, w*128+128). Batch tiled by 16.
// h tiles live in LDS (not VGPRs) to keep the wave under 256 registers.
// ---------------------------------------------------------------------------
__global__ __launch_bounds__(256) void particle_grad(
    const float* __restrict__ W1, const float* __restrict__ b1,
    const float* __restrict__ W2, const float* __restrict__ b2,
    const float* __restrict__ X, const float* __restrict__ y,
    float* __restrict__ dlog) {
  const int p = blockIdx.x;
  const int tid = threadIdx.x;
  const int wave = tid >> 5;
  const int lane = tid & 31;
  const int ln = lane & 15;            // N (or M) index within tile
  const int kk = (lane >> 4) << 1;     // K pair base: 0 or 2
  const int mhalf = (lane >> 4) << 3;  // C-frag row offset: 0 or 8

  __shared__ float b1s[HID];
  __shared__ float w2s[HID];
  __shared__ float wsum[8][16];
  __shared__ float rbuf[16];
  __shared__ float hs[8][8 * HT_STRIDE];  // per-wave 8 h tiles, 17-padded rows
  __shared__ float dhT[8][HT_STRIDE];     // per-wave transposed dh staging

  const float* W1p = W1 + (size_t)p * (HID * DIN);
  for (int i = tid; i < HID; i += 256) {
    b1s[i] = b1[(size_t)p * HID + i];
    w2s[i] = W2[(size_t)p * HID + i];
  }
  __syncthreads();

  v8f gw1[8];
  float gw2a[8], gb1a[8];
  for (int t = 0; t < 8; ++t) { gw1[t] = vzero8(); gw2a[t] = 0.0f; gb1a[t] = 0.0f; }
  float gb2a = 0.0f;
  const int j0w = wave * 128;
  const float b2p = b2[p];

  for (int b0 = 0; b0 < BATCH; b0 += 16) {
    // ---- A fragments from X tile: A[m=batch_row, k=din] --------------------
    v2f afr[4];
    {
      const float2* X2 = (const float2*)(X + (size_t)(b0 + ln) * DIN);
      for (int q = 0; q < 4; ++q) {
        float2 t2 = X2[(q * 4 + kk) >> 1];
        afr[q].x = t2.x; afr[q].y = t2.y;
      }
    }
    // ---- forward: h = relu(X @ W1^T + b1); fuse yhat partial; h -> LDS -----
    v8f part = vzero8();
    for (int t = 0; t < 8; ++t) {
      const int j0 = j0w + t * 16;
      v8f c = vzero8();
      const float2* Wrow = (const float2*)(W1p + (size_t)(j0 + ln) * DIN);
      for (int q = 0; q < 4; ++q) {
        float2 t2 = Wrow[(q * 4 + kk) >> 1];   // B[k,n] = W1[j0+n, k] ("[N,K]" load)
        v2f bf; bf.x = t2.x; bf.y = t2.y;
        c = wmma4(afr[q], bf, c);
      }
      const float bb = b1s[j0 + ln];
      const float w2c = w2s[j0 + ln];
      float* ht = &hs[wave][t * HT_STRIDE];
      for (int v = 0; v < 8; ++v) {
        float hv = c[v] + bb;
        hv = hv > 0.0f ? hv : 0.0f;
        part[v] += hv * w2c;
        ht[(mhalf + v) * 17 + ln] = hv;
      }
    }
    // ---- yhat row sums: deterministic xor-shuffle + cross-wave LDS ---------
    for (int off = 1; off <= 8; off <<= 1)
      for (int v = 0; v < 8; ++v) part[v] += __shfl_xor(part[v], off, 32);
    if (ln == 0)
      for (int v = 0; v < 8; ++v) wsum[wave][mhalf + v] = part[v];
    __syncthreads();
    if (tid < 16) {
      float ys = 0.0f;
      for (int w = 0; w < 8; ++w) ys += wsum[w][tid];
      rbuf[tid] = y[b0 + tid] - (ys + b2p);   // r = y - yhat
    }
    __syncthreads();
    float rv[8];
    for (int v = 0; v < 8; ++v) rv[v] = rbuf[mhalf + v];
    if (tid == 0)
      for (int rr = 0; rr < 16; ++rr) gb2a += rbuf[rr];

    // ---- grads: gW2 += r^T h ; dh = (h>0)*r*w2 ; gb1 += sum dh ; gW1 += dh^T X
    float* dT = dhT[wave];
    for (int t = 0; t < 8; ++t) {
      const float w2c = w2s[j0w + t * 16 + ln];
      const float* ht = &hs[wave][t * HT_STRIDE];
      for (int v = 0; v < 8; ++v) {
        const float hv = ht[(mhalf + v) * 17 + ln];
        gw2a[t] += rv[v] * hv;
        const float d = (hv > 0.0f) ? rv[v] * w2c : 0.0f;
        gb1a[t] += d;
        dT[ln * 17 + (mhalf + v)] = d;        // store already transposed
      }
      for (int q = 0; q < 4; ++q) {
        v2f aa, bb;                           // A[m=jloc,k=row] ; B[k=row,n=din]
        aa.x = dT[ln * 17 + q * 4 + kk];
        aa.y = dT[ln * 17 + q * 4 + kk + 1];
        bb.x = X[(size_t)(b0 + q * 4 + kk) * DIN + ln];
        bb.y = X[(size_t)(b0 + q * 4 + kk + 1) * DIN + ln];
        gw1[t] = wmma4(aa, bb, gw1[t]);
      }
    }
    __syncthreads();
  }

  // ---- writeout (subtract Gaussian-prior gradient: -param) -----------------
  float* dlp = dlog + (size_t)p * D_PAD;
  for (int t = 0; t < 8; ++t) {
    const int j0 = j0w + t * 16;
    for (int v = 0; v < 8; ++v) {
      const int j = j0 + mhalf + v;
      dlp[j * DIN + ln] = gw1[t][v] - W1p[(size_t)j * DIN + ln];
    }
    const float g2 = gw2a[t] + __shfl_xor(gw2a[t], 16, 32);
    const float g1 = gb1a[t] + __shfl_xor(gb1a[t], 16, 32);
    if (lane < 16) {
      const int j = j0 + ln;
      dlp[OFF_B1 + j] = g1 - b1s[j];
      dlp[OFF_W2 + j] = g2 - w2s[j];
    }
  }
  if (tid == 0) dlp[OFF_B2] = gb2a - b2p;
  if (tid >= 1 && tid < 16) dlp[OFF_B2 + tid] = 0.0f;   // zero padding
}

// ---------------------------------------------------------------------------
// K2: Gram = theta @ theta^T via WMMA; kd = exp(-max(d2,0)/2); s = row sums.
// One block, 8 waves, 2 of the 16 output tiles per wave.
// ---------------------------------------------------------------------------
__global__ __launch_bounds__(256) void rbf_kernel(const float* __restrict__ theta,
                                                  float* __restrict__ kd,
                                                  float* __restrict__ s) {
  const int tid = threadIdx.x;
  const int wave = tid >> 5, lane = tid & 31;
  const int ln = lane & 15, kk = (lane >> 4) << 1, mhalf = (lane >> 4) << 3;
  __shared__ float gram[64 * 64];
  __shared__ float kds[64 * 64];

  v8f acc[2] = {vzero8(), vzero8()};
  const int t0 = wave * 2, t1 = wave * 2 + 1;
  const int mt0 = t0 >> 2, nt0 = t0 & 3, mt1 = t1 >> 2, nt1 = t1 & 3;
  const float2* Am0 = (const float2*)(theta + (size_t)(mt0 * 16 + ln) * D_PAD);
  const float2* Bn0 = (const float2*)(theta + (size_t)(nt0 * 16 + ln) * D_PAD);
  const float2* Am1 = (const float2*)(theta + (size_t)(mt1 * 16 + ln) * D_PAD);
  const float2* Bn1 = (const float2*)(theta + (size_t)(nt1 * 16 + ln) * D_PAD);
  for (int k0 = 0; k0 < D_PAD; k0 += 4) {
    const int ki = (k0 + kk) >> 1;
    v2f a, b;
    float2 t2;
    t2 = Am0[ki]; a.x = t2.x; a.y = t2.y;
    t2 = Bn0[ki]; b.x = t2.x; b.y = t2.y;
    acc[0] = wmma4(a, b, acc[0]);
    t2 = Am1[ki]; a.x = t2.x; a.y = t2.y;
    t2 = Bn1[ki]; b.x = t2.x; b.y = t2.y;
    acc[1] = wmma4(a, b, acc[1]);
  }
  for (int v = 0; v < 8; ++v) {
    gram[(mt0 * 16 + mhalf + v) * 64 + nt0 * 16 + ln] = acc[0][v];
    gram[(mt1 * 16 + mhalf + v) * 64 + nt1 * 16 + ln] = acc[1][v];
  }
  __syncthreads();
  for (int idx = tid; idx < 4096; idx += 256) {
    const int i = idx >> 6, j = idx & 63;
    float d2 = gram[i * 64 + i] + gram[j * 64 + j] - 2.0f * gram[idx];
    d2 = d2 > 0.0f ? d2 : 0.0f;
    kds[idx] = __expf(-0.5f * d2);
  }
  __syncthreads();
  if (tid < 64) {
    float a = 0.0f;
    for (int j = 0; j < 64; ++j) a += kds[tid * 64 + j];   // fixed-order sum
    s[tid] = a;
  }
  for (int idx = tid; idx < 4096; idx += 256) kd[idx] = kds[idx];
}

// ---------------------------------------------------------------------------
// K3: out = theta + c0*( kd @ (dlog - theta) + s .* theta )   (rbf^2 == 1)
// One 16-col n-tile per wave; 4 m-tiles (64 particles) accumulated per wave.
// ---------------------------------------------------------------------------
__global__ __launch_bounds__(128) void svgd_update(
    const float* __restrict__ theta, const float* __restrict__ dlog,
    const float* __restrict__ kd, const float* __restrict__ s,
    float* __restrict__ out) {
  const int tid = threadIdx.x;
  const int wave = tid >> 5, lane = tid & 31;
  const int ln = lane & 15, kk = (lane >> 4) << 1, mhalf = (lane >> 4) << 3;
  __shared__ float kds[64 * 64];
  __shared__ float ss[64];
  for (int i = tid; i < 4096; i += 128) kds[i] = kd[i];
  if (tid < 64) ss[tid] = s[tid];
  __syncthreads();

  const int nt = blockIdx.x * 4 + wave;
  if (nt >= NTILES) return;               // wave-uniform exit (EXEC stays full)
  const int col = nt * 16 + ln;

  v8f acc[4];
  for (int m = 0; m < 4; ++m) acc[m] = vzero8();
  for (int k0 = 0; k0 < 64; k0 += 4) {
    v2f b;                                 // B[k=particle j, n=col]
    const size_t i0 = (size_t)(k0 + kk) * D_PAD + col;
    b.x = dlog[i0] - theta[i0];
    b.y = dlog[i0 + D_PAD] - theta[i0 + D_PAD];
    for (int mt = 0; mt < 4; ++mt) {
      v2f a;                               // A[m=particle i, k=particle j]
      a.x = kds[(mt * 16 + ln) * 64 + k0 + kk];
      a.y = kds[(mt * 16 + ln) * 64 + k0 + kk + 1];
      acc[mt] = wmma4(a, b, acc[mt]);
    }
  }
  if (col < D_FLAT) {
    for (int mt = 0; mt < 4; ++mt) {
      for (int v = 0; v < 8; ++v) {
        const int i = mt * 16 + mhalf + v;
        const float th = theta[(size_t)i * D_PAD + col];
        out[(size_t)i * D_FLAT + col] = th + STEP_OVER_N * (acc[mt][v] + ss[i] * th);
      }
    }
  }
}

// ---------------------------------------------------------------------------
extern "C" void kernel_launch(void* const* d_in, const int* in_sizes, int n_in,
                              void* d_out, int out_size, void* d_ws, size_t ws_size,
                              hipStream_t stream) {
  const float* W1 = (const float*)d_in[0];
  const float* b1 = (const float*)d_in[1];
  const float* W2 = (const float*)d_in[2];
  const float* b2 = (const float*)d_in[3];
  const float* X  = (const float*)d_in[4];
  const float* y  = (const float*)d_in[5];
  float* out = (float*)d_out;

  float* theta = (float*)d_ws;                          // 64 * 18448
  float* dlog  = theta + (size_t)N_PART * D_PAD;        // 64 * 18448
  float* kd    = dlog  + (size_t)N_PART * D_PAD;        // 64 * 64
  float* s     = kd + 64 * 64;                          // 64

  pack_theta<<<256, 256, 0, stream>>>(W1, b1, W2, b2, theta);
  particle_grad<<<N_PART, 256, 0, stream>>>(W1, b1, W2, b2, X, y, dlog);
  rbf_kernel<<<1, 256, 0, stream>>>(theta, kd, s);
  svgd_update<<<(NTILES + 3) / 4, 128, 0, stream>>>(theta, dlog, kd, s, out);
}